// GNNSAGEConv_53953379173287
// MI455X (gfx1250) — compile-verified
//
#include <hip/hip_runtime.h>
#include <cstddef>

typedef __attribute__((ext_vector_type(2))) float v2f;
typedef __attribute__((ext_vector_type(8))) float v8f;

// ---------------------------------------------------------------------------
// Degree count: one thread per edge, f32 atomic into cnt[dst[e]].
// ---------------------------------------------------------------------------
__global__ void degree_kernel(const int* __restrict__ dst,
                              float* __restrict__ cnt, int nedges) {
    int e = blockIdx.x * blockDim.x + threadIdx.x;
    if (e < nedges) {
        unsafeAtomicAdd(&cnt[dst[e]], 1.0f);   // global_atomic_add_f32
    }
}

// ---------------------------------------------------------------------------
// Edge scatter-add: one wave32 per edge. Lane L moves features L, L+32, ...
// Coalesced row gather from x[src], L2-resident f32 atomics into sum[dst].
// ---------------------------------------------------------------------------
template <int F>
__global__ void scatter_add_kernel(const float* __restrict__ x,
                                   const int* __restrict__ src,
                                   const int* __restrict__ dst,
                                   float* __restrict__ sum, int nedges) {
    int gid  = blockIdx.x * blockDim.x + threadIdx.x;
    int e    = gid >> 5;
    int lane = gid & 31;
    if (e >= nedges) return;
    int s = src[e];
    int d = dst[e];
    const float* xs = x + (size_t)s * F;
    float*       sd = sum + (size_t)d * F;
#pragma unroll
    for (int j = 0; j < F / 32; ++j) {
        float v = xs[lane + 32 * j];
        unsafeAtomicAdd(sd + lane + 32 * j, v);
    }
}

// ---------------------------------------------------------------------------
// Fused SAGE dense stage:
//   out[i,:] = act( (sum[i,:]/max(cnt[i],1)) @ Wl^T + bl + xin[i,:] @ Wr^T )
//
// Grid: N/16 blocks of 128 threads (4 waves). Each wave computes NT 16x16
// output tiles with V_WMMA_F32_16X16X4_F32, chaining K in steps of 4.
//
// Fragment layouts (CDNA5 ISA 7.12.2, wave32):
//   A (16x4 f32, 2 VGPRs): lanes 0-15 row M=lane hold K={0,1};
//                          lanes 16-31 row M=lane-16 hold K={2,3}.
//   B (4x16 f32, 2 VGPRs): lanes 0-15 col N=lane hold K={0,1};
//                          lanes 16-31 col N=lane-16 hold K={2,3}.
//   C/D (16x16 f32, 8 VGPRs): VGPR r -> M = r + (lane<16 ? 0 : 8), N = lane&15.
// ---------------------------------------------------------------------------
template <int INC, int OUTC, int NT, bool RELU>
__global__ void sage_layer_kernel(const float* __restrict__ sum,
                                  const float* __restrict__ cnt,
                                  const float* __restrict__ xin,
                                  const float* __restrict__ Wl,   // [OUTC][INC]
                                  const float* __restrict__ bl,   // [OUTC]
                                  const float* __restrict__ Wr,   // [OUTC][INC]
                                  float* __restrict__ out,        // [N][OUTC]
                                  int nnodes) {
    const int lane    = threadIdx.x & 31;
    const int wave    = threadIdx.x >> 5;
    const int r0      = blockIdx.x * 16;
    int row           = r0 + (lane & 15);
    if (row >= nnodes) row = nnodes - 1;       // never triggers for N%16==0
    const int koff    = (lane < 16) ? 0 : 2;   // K-pair selector for A and B
    const int halfsel = (lane < 16) ? 0 : 8;   // M offset for C/D rows

    const float rcnt = 1.0f / fmaxf(cnt[row], 1.0f);

    const float* arow_s = sum + (size_t)row * INC;
    const float* arow_x = xin + (size_t)row * INC;

    for (int t = 0; t < NT; ++t) {
        const int n0  = (wave * NT + t) * 16;
        const int col = n0 + (lane & 15);

        const float bias = bl[col];
        v8f acc;
#pragma unroll
        for (int r = 0; r < 8; ++r) acc[r] = bias;

        const float* brow_l = Wl + (size_t)col * INC;
        const float* brow_r = Wr + (size_t)col * INC;

#pragma unroll
        for (int kk = 0; kk < INC / 4; ++kk) {
            const int kb = 4 * kk + koff;            // even -> 8B aligned
            v2f am = *(const v2f*)(arow_s + kb);
            am *= rcnt;                               // neighbor mean
            v2f av = *(const v2f*)(arow_x + kb);
            v2f wl = *(const v2f*)(brow_l + kb);      // B[k][n] = Wl[n][k]
            v2f wr = *(const v2f*)(brow_r + kb);
            acc = __builtin_amdgcn_wmma_f32_16x16x4_f32(
                false, am, false, wl, (short)0, acc, false, false);
            acc = __builtin_amdgcn_wmma_f32_16x16x4_f32(
                false, av, false, wr, (short)0, acc, false, false);
        }

#pragma unroll
        for (int r = 0; r < 8; ++r) {
            float v = acc[r];
            if (RELU) v = fmaxf(v, 0.0f);
            int orow = r0 + halfsel + r;
            if (orow < nnodes) out[(size_t)orow * OUTC + col] = v;
        }
    }
}

// ---------------------------------------------------------------------------
// Host launcher
// ---------------------------------------------------------------------------
extern "C" void kernel_launch(void* const* d_in, const int* in_sizes, int n_in,
                              void* d_out, int out_size, void* d_ws, size_t ws_size,
                              hipStream_t stream) {
    const float* x   = (const float*)d_in[0];
    const int*   ei  = (const int*)d_in[1];
    const float* W1l = (const float*)d_in[2];
    const float* b1l = (const float*)d_in[3];
    const float* W1r = (const float*)d_in[4];
    const float* W2l = (const float*)d_in[5];
    const float* b2l = (const float*)d_in[6];
    const float* W2r = (const float*)d_in[7];
    float* out = (float*)d_out;

    const int Nn = in_sizes[0] / 64;   // 100000 nodes (H = 64)
    const int Ee = in_sizes[1] / 2;    // 1600000 edges
    const int* src = ei;               // edge_index[0]
    const int* dst = ei + Ee;          // edge_index[1]

    // Workspace layout: cnt[N] | sum[N*128] (reused) | h[N*128]
    float* cnt  = (float*)d_ws;
    float* sums = cnt + Nn;
    float* h    = sums + (size_t)Nn * 128;

    const int mblocks = (Nn + 15) / 16;     // 6250
    const long long sth = (long long)Ee * 32;
    const int sblocks = (int)((sth + 255) / 256);

    // ---- Layer 1: H=64 -> 128, ReLU ----
    hipMemsetAsync(cnt, 0, (size_t)Nn * sizeof(float), stream);
    hipMemsetAsync(sums, 0, (size_t)Nn * 64 * sizeof(float), stream);

    degree_kernel<<<(Ee + 255) / 256, 256, 0, stream>>>(dst, cnt, Ee);
    scatter_add_kernel<64><<<sblocks, 256, 0, stream>>>(x, src, dst, sums, Ee);
    sage_layer_kernel<64, 128, 2, true><<<mblocks, 128, 0, stream>>>(
        sums, cnt, x, W1l, b1l, W1r, h, Nn);

    // ---- Layer 2: 128 -> 64 ----
    hipMemsetAsync(sums, 0, (size_t)Nn * 128 * sizeof(float), stream);
    scatter_add_kernel<128><<<sblocks, 256, 0, stream>>>(h, src, dst, sums, Ee);
    sage_layer_kernel<128, 64, 1, false><<<mblocks, 128, 0, stream>>>(
        sums, cnt, h, W2l, b2l, W2r, out, Nn);
}